// Embedding_78666620994160
// MI455X (gfx1250) — compile-verified
//
#include <hip/hip_runtime.h>
#include <stdint.h>

#define VOCAB 50257
#define EMB   512

// sqrt(512)
constexpr float SCALE = 22.627416997969522f;

typedef float v4f __attribute__((ext_vector_type(4)));

// ---------------------------------------------------------------------------
// Phase 1: tiled transpose  W (EMB x VOCAB, row-major) -> Wt (VOCAB x EMB)
// 32x32 tiles through LDS (padded), coalesced on both sides.
// W is read exactly once -> non-temporal loads (don't evict Wt from L2).
// Wt is re-read heavily by phase 2 -> regular-temporal stores (stay in L2).
// ---------------------------------------------------------------------------
__global__ __launch_bounds__(256)
void transpose_w(const float* __restrict__ W, float* __restrict__ Wt) {
  __shared__ float tile[32][33];          // +1 pad: no LDS bank conflicts
  const int v0 = blockIdx.x * 32;         // vocab tile origin
  const int e0 = blockIdx.y * 32;         // emb tile origin
  const int tx = threadIdx.x;             // 0..31
  const int ty = threadIdx.y;             // 0..7

#pragma unroll
  for (int j = 0; j < 4; ++j) {
    const int e = e0 + ty + 8 * j;        // EMB==512 is a multiple of 32
    const int v = v0 + tx;
    float val = 0.0f;
    if (v < VOCAB) val = __builtin_nontemporal_load(&W[(size_t)e * VOCAB + v]);
    tile[ty + 8 * j][tx] = val;
  }
  __syncthreads();
#pragma unroll
  for (int j = 0; j < 4; ++j) {
    const int v = v0 + ty + 8 * j;
    if (v < VOCAB) Wt[(size_t)v * EMB + (e0 + tx)] = tile[tx][ty + 8 * j];
  }
}

// ---------------------------------------------------------------------------
// Phase 2: gather rows of Wt via async global->LDS DMA (gfx1250 path),
// double-buffered per wave, fused (x + b) * sqrt(EMB).
// Per wave, per token: 4x global_load_async_to_lds_b128 = one 2KB row.
// ASYNCcnt completes in order -> after issuing the next row's 4 ops, waiting
// asynccnt<=4 guarantees the current row has fully landed in LDS.
// Output is write-once -> non-temporal b128 stores (keep Wt resident in L2).
// ---------------------------------------------------------------------------
#define WAIT_ASYNCCNT(N) asm volatile("s_wait_asynccnt %0" ::"n"(N) : "memory")

__global__ __launch_bounds__(256)
void gather_async(const int* __restrict__ ids, const float* __restrict__ Wt,
                  const float* __restrict__ bias, float* __restrict__ out,
                  int ntok) {
  // 8 waves/block * 2 buffers * 128 v4f (2KB) = 32KB LDS
  __shared__ __align__(16) v4f smem[8 * 256];

  const int lane = threadIdx.x & 31;
  const int wave = threadIdx.x >> 5;
  const int gw   = blockIdx.x * 8 + wave;   // global wave id
  const int nw   = gridDim.x * 8;           // total waves

  // Preload bias: lane owns e = k*128 + lane*4 .. +3 for k=0..3
  const v4f* b4 = (const v4f*)bias;
  v4f bk[4];
#pragma unroll
  for (int k = 0; k < 4; ++k) bk[k] = b4[k * 32 + lane];

  v4f* wbase = &smem[wave * 256];
  v4f* out4  = (v4f*)out;

  auto issue = [&](int token, int buf) {
    const char* src =
        (const char*)(Wt + (size_t)ids[token] * EMB) + lane * 16;
    // low 32 bits of a flat LDS pointer == LDS byte offset (aperture in hi32)
    const unsigned ldsa = (unsigned)(uintptr_t)&wbase[buf * 128 + lane];
#pragma unroll
    for (int k = 0; k < 4; ++k) {
      asm volatile("global_load_async_to_lds_b128 %0, %1, off"
                   ::"v"(ldsa + (unsigned)(k * 512)), "v"(src + k * 512)
                   : "memory");
    }
  };

  int t = gw;
  if (t < ntok) issue(t, 0);
  int i = 0;
  while (t < ntok) {
    const int tn = t + nw;
    if (tn < ntok) {
      issue(tn, (i + 1) & 1);   // prefetch next row into the other buffer
      WAIT_ASYNCCNT(4);         // in-order: current row (first 4 ops) done
    } else {
      WAIT_ASYNCCNT(0);
    }
    const v4f* lb = &wbase[(i & 1) * 128];
    const size_t ob = (size_t)t * 128;
#pragma unroll
    for (int k = 0; k < 4; ++k) {
      const v4f w = lb[k * 32 + lane];               // ds_load_b128
      const v4f r = (w + bk[k]) * SCALE;
      __builtin_nontemporal_store(r, &out4[ob + k * 32 + lane]);
    }
    t = tn;
    ++i;
  }
}

// ---------------------------------------------------------------------------
// Fallback if workspace can't hold Wt (103 MB): direct strided gather.
// ---------------------------------------------------------------------------
__global__ __launch_bounds__(256)
void gather_direct(const int* __restrict__ ids, const float* __restrict__ W,
                   const float* __restrict__ bias, float* __restrict__ out,
                   int ntok) {
  const int gid = blockIdx.x * blockDim.x + threadIdx.x;
  if (gid >= ntok * 128) return;
  const int token = gid >> 7;
  const int q     = gid & 127;      // v4f index within the row
  const int id    = ids[token];
  const v4f bb = ((const v4f*)bias)[q];
  const int e = q * 4;
  v4f r;
  r.x = (W[(size_t)(e + 0) * VOCAB + id] + bb.x) * SCALE;
  r.y = (W[(size_t)(e + 1) * VOCAB + id] + bb.y) * SCALE;
  r.z = (W[(size_t)(e + 2) * VOCAB + id] + bb.z) * SCALE;
  r.w = (W[(size_t)(e + 3) * VOCAB + id] + bb.w) * SCALE;
  __builtin_nontemporal_store(r, &((v4f*)out)[gid]);
}

// ---------------------------------------------------------------------------
extern "C" void kernel_launch(void* const* d_in, const int* in_sizes, int n_in,
                              void* d_out, int out_size, void* d_ws,
                              size_t ws_size, hipStream_t stream) {
  const int*   ids = (const int*)d_in[0];     // (8*4096) int32 token ids
  const float* W   = (const float*)d_in[1];   // (EMB, VOCAB) f32
  const float* b   = (const float*)d_in[2];   // (EMB,) f32
  float*       out = (float*)d_out;           // (ntok, EMB) f32
  const int ntok = in_sizes[0];

  const size_t need = (size_t)VOCAB * EMB * sizeof(float);
  if (ws_size >= need) {
    float* Wt = (float*)d_ws;
    dim3 tb(32, 8);
    dim3 tg((VOCAB + 31) / 32, EMB / 32);
    hipLaunchKernelGGL(transpose_w, tg, tb, 0, stream, W, Wt);
    hipLaunchKernelGGL(gather_async, dim3(256), dim3(256), 0, stream,
                       ids, Wt, b, out, ntok);
  } else {
    const int total  = ntok * 128;
    const int blocks = (total + 255) / 256;
    hipLaunchKernelGGL(gather_direct, dim3(blocks), dim3(256), 0, stream,
                       ids, W, b, out, ntok);
  }
}